// SolveNewsvendor_86260123174049
// MI455X (gfx1250) — compile-verified
//
#include <hip/hip_runtime.h>
#include <stdint.h>

// Batched newsvendor QP interior-point solver for MI455X (gfx1250, wave32).
// One sample per wave32; each lane owns scenario indices k=lane and k=lane+32.
// The KKT system is an arrow matrix -> exact O(K) Schur-complement solve,
// replacing the reference's dense 129x129 LU (O(N^3)) per sample per iter.
// y[b,:] is staged through the CDNA5 async global->LDS path (ASYNCcnt).

#define KDIM 64
#define NITER_IP 30
#define SIGMA_IP 0.1
#define BIG_IP 1e10
#define WAVES_PER_BLOCK 8

__device__ __forceinline__ double shfl_xor_f64(double v, int m) {
  union { double d; int i[2]; } a;
  a.d = v;
  a.i[0] = __shfl_xor(a.i[0], m, 32);
  a.i[1] = __shfl_xor(a.i[1], m, 32);
  return a.d;
}

__device__ __forceinline__ double wave_sum(double v) {
#pragma unroll
  for (int m = 16; m; m >>= 1) v += shfl_xor_f64(v, m);
  return v;
}

__device__ __forceinline__ double wave_min(double v) {
#pragma unroll
  for (int m = 16; m; m >>= 1) v = fmin(v, shfl_xor_f64(v, m));
  return v;
}

__device__ __forceinline__ double wave_max(double v) {
#pragma unroll
  for (int m = 16; m; m >>= 1) v = fmax(v, shfl_xor_f64(v, m));
  return v;
}

__global__ __launch_bounds__(256, 1) void newsvendor_ip_kernel(
    const float* __restrict__ y, const double* __restrict__ qb,
    const double* __restrict__ pb, const double* __restrict__ dv,
    float* __restrict__ out, int B) {
  __shared__ float ylds[WAVES_PER_BLOCK][KDIM];
  const int lane = threadIdx.x & 31;
  const int wv = threadIdx.x >> 5;
  const int bid = blockIdx.x * WAVES_PER_BLOCK + wv;
  if (bid >= B) return;  // wave-coherent exit

  // ---- CDNA5 async global->LDS staging of y[bid, 0:64] (ASYNCcnt path) ----
  {
    const float* g = y + (size_t)bid * KDIM;
    unsigned long long g0 = (unsigned long long)(uintptr_t)(g + lane);
    unsigned long long g1 = (unsigned long long)(uintptr_t)(g + lane + 32);
    unsigned int l0 = (unsigned int)(uintptr_t)&ylds[wv][lane];
    unsigned int l1 = (unsigned int)(uintptr_t)&ylds[wv][lane + 32];
    asm volatile("global_load_async_to_lds_b32 %0, %1, off"
                 :: "v"(l0), "v"(g0) : "memory");
    asm volatile("global_load_async_to_lds_b32 %0, %1, off"
                 :: "v"(l1), "v"(g1) : "memory");
    asm volatile("s_wait_asynccnt 0" ::: "memory");
  }
  const double y0 = (double)ylds[wv][lane];
  const double y1 = (double)ylds[wv][lane + 32];

  // ---- per-lane constants (k = lane, lane+32) ----
  const double d0 = dv[lane], d1 = dv[lane + 32];
  const double q0c = qb[0], p0c = pb[0];
  const double qq0 = qb[1 + lane] * y0, qq1 = qb[1 + lane + 32] * y1;
  const double qr0 = qb[1 + KDIM + lane] * y0, qr1 = qb[1 + KDIM + lane + 32] * y1;
  const double pq0 = pb[1 + lane] * y0, pq1 = pb[1 + lane + 32] * y1;
  const double pr0 = pb[1 + KDIM + lane] * y0, pr1 = pb[1 + KDIM + lane + 32] * y1;

  // strictly feasible start: zc = max(d)+1
  const double zc = wave_max(fmax(d0, d1)) + 1.0;
  double z0 = zc;
  double zq0 = 1.0, zq1 = 1.0;
  double zr0 = zc - d0 + 1.0, zr1 = zc - d1 + 1.0;
  double l10 = 1.0, l11 = 1.0, l20 = 1.0, l21 = 1.0;
  double l3c = 1.0, l3q0 = 1.0, l3q1 = 1.0, l3r0 = 1.0, l3r1 = 1.0;

  for (int it = 0; it < NITER_IP; ++it) {
    // slacks: s1_k = z0 + q_k - d_k ; s2_k = d_k - z0 + r_k ; s3 = z
    const double s10 = z0 + zq0 - d0, s11 = z0 + zq1 - d1;
    const double s20 = d0 - z0 + zr0, s21 = d1 - z0 + zr1;

    // reductions: sum(lam2 - lam1) and mu = mean(lam*s) over M = 257
    double sumL = (l20 - l10) + (l21 - l11);
    double mup = l10 * s10 + l11 * s11 + l20 * s20 + l21 * s21 +
                 l3q0 * zq0 + l3q1 * zq1 + l3r0 * zr0 + l3r1 * zr1;
    sumL = wave_sum(sumL);
    const double mu = (wave_sum(mup) + l3c * z0) * (1.0 / 257.0);
    const double sm = SIGMA_IP * mu;

    // dual residual r_dual = q.*z + p + G'lam
    const double rd0 = q0c * z0 + p0c + sumL - l3c;
    const double rdq0 = qq0 * zq0 + pq0 - l10 - l3q0;
    const double rdq1 = qq1 * zq1 + pq1 - l11 - l3q1;
    const double rdr0 = qr0 * zr0 + pr0 - l20 - l3r0;
    const double rdr1 = qr1 * zr1 + pr1 - l21 - l3r1;

    // reciprocals of slacks
    const double is10 = 1.0 / s10, is11 = 1.0 / s11;
    const double is20 = 1.0 / s20, is21 = 1.0 / s21;
    const double izq0 = 1.0 / zq0, izq1 = 1.0 / zq1;
    const double izr0 = 1.0 / zr0, izr1 = 1.0 / zr1;
    const double iz0 = 1.0 / z0;

    // centering residuals and D = lam/s
    const double rc10 = l10 * s10 - sm, rc11 = l11 * s11 - sm;
    const double rc20 = l20 * s20 - sm, rc21 = l21 * s21 - sm;
    const double rc3c = l3c * z0 - sm;
    const double rc3q0 = l3q0 * zq0 - sm, rc3q1 = l3q1 * zq1 - sm;
    const double rc3r0 = l3r0 * zr0 - sm, rc3r1 = l3r1 * zr1 - sm;
    const double D10 = l10 * is10, D11 = l11 * is11;
    const double D20 = l20 * is20, D21 = l21 * is21;
    const double D3c = l3c * iz0;
    const double D3q0 = l3q0 * izq0, D3q1 = l3q1 * izq1;
    const double D3r0 = l3r0 * izr0, D3r1 = l3r1 * izr1;

    const double t10 = rc10 * is10, t11 = rc11 * is11;
    const double t20 = rc20 * is20, t21 = rc21 * is21;
    const double t3c = rc3c * iz0;
    const double t3q0 = rc3q0 * izq0, t3q1 = rc3q1 * izq1;
    const double t3r0 = rc3r0 * izr0, t3r1 = rc3r1 * izr1;

    // rhs = -r_dual + (r_cent/s)G
    const double rhsq0 = -rdq0 - t10 - t3q0, rhsq1 = -rdq1 - t11 - t3q1;
    const double rhsr0 = -rdr0 - t20 - t3r0, rhsr1 = -rdr1 - t21 - t3r1;

    // arrow matrix: pivot a, diagonals bq/cr, couplings u = D1, v = -D2
    const double bq0 = qq0 + D10 + D3q0, bq1 = qq1 + D11 + D3q1;
    const double cr0 = qr0 + D20 + D3r0, cr1 = qr1 + D21 + D3r1;
    const double ib0 = 1.0 / bq0, ib1 = 1.0 / bq1;
    const double ic0 = 1.0 / cr0, ic1 = 1.0 / cr1;
    const double u0 = D10, u1 = D11, v0 = -D20, v1 = -D21;

    double apart = D10 + D11 + D20 + D21;
    double r0part = -t10 - t11 + t20 + t21;
    double sc1 = u0 * u0 * ib0 + u1 * u1 * ib1 + v0 * v0 * ic0 + v1 * v1 * ic1;
    double sc2 = u0 * rhsq0 * ib0 + u1 * rhsq1 * ib1 +
                 v0 * rhsr0 * ic0 + v1 * rhsr1 * ic1;
    apart = wave_sum(apart);
    r0part = wave_sum(r0part);
    sc1 = wave_sum(sc1);
    sc2 = wave_sum(sc2);

    // Schur complement on the scalar pivot
    const double a = q0c + apart + D3c;
    const double rhs0 = -rd0 + r0part - t3c;
    const double dz0 = (rhs0 - sc2) / (a - sc1);
    const double dzq0 = (rhsq0 - u0 * dz0) * ib0, dzq1 = (rhsq1 - u1 * dz0) * ib1;
    const double dzr0 = (rhsr0 - v0 * dz0) * ic0, dzr1 = (rhsr1 - v1 * dz0) * ic1;

    // ds = -G dz
    const double ds10 = dz0 + dzq0, ds11 = dz0 + dzq1;
    const double ds20 = -dz0 + dzr0, ds21 = -dz0 + dzr1;

    // dlam = (-r_cent - lam.*ds)/s
    const double dl10 = (-rc10 - l10 * ds10) * is10;
    const double dl11 = (-rc11 - l11 * ds11) * is11;
    const double dl20 = (-rc20 - l20 * ds20) * is20;
    const double dl21 = (-rc21 - l21 * ds21) * is21;
    const double dl3c = (-rc3c - l3c * dz0) * iz0;
    const double dl3q0 = (-rc3q0 - l3q0 * dzq0) * izq0;
    const double dl3q1 = (-rc3q1 - l3q1 * dzq1) * izq1;
    const double dl3r0 = (-rc3r0 - l3r0 * dzr0) * izr0;
    const double dl3r1 = (-rc3r1 - l3r1 * dzr1) * izr1;

    // fraction-to-boundary step
    double am = BIG_IP;
#define STEPC(sv, dsv) am = fmin(am, ((dsv) < 0.0) ? (-(sv) / (dsv)) : BIG_IP)
    STEPC(s10, ds10); STEPC(s11, ds11); STEPC(s20, ds20); STEPC(s21, ds21);
    STEPC(z0, dz0); STEPC(zq0, dzq0); STEPC(zq1, dzq1);
    STEPC(zr0, dzr0); STEPC(zr1, dzr1);
    STEPC(l10, dl10); STEPC(l11, dl11); STEPC(l20, dl20); STEPC(l21, dl21);
    STEPC(l3c, dl3c); STEPC(l3q0, dl3q0); STEPC(l3q1, dl3q1);
    STEPC(l3r0, dl3r0); STEPC(l3r1, dl3r1);
#undef STEPC
    am = wave_min(am);
    const double alpha = fmin(1.0, 0.99 * am);

    // update iterate
    z0 += alpha * dz0;
    zq0 += alpha * dzq0; zq1 += alpha * dzq1;
    zr0 += alpha * dzr0; zr1 += alpha * dzr1;
    l10 += alpha * dl10; l11 += alpha * dl11;
    l20 += alpha * dl20; l21 += alpha * dl21;
    l3c += alpha * dl3c;
    l3q0 += alpha * dl3q0; l3q1 += alpha * dl3q1;
    l3r0 += alpha * dl3r0; l3r1 += alpha * dl3r1;
  }

  if (lane == 0) out[bid] = (float)z0;
}

extern "C" void kernel_launch(void* const* d_in, const int* in_sizes, int n_in,
                              void* d_out, int out_size, void* d_ws, size_t ws_size,
                              hipStream_t stream) {
  (void)n_in; (void)d_ws; (void)ws_size; (void)out_size;
  const float* y = (const float*)d_in[0];     // (B, K) float32
  const double* qb = (const double*)d_in[1];  // (N,) float64
  const double* pb = (const double*)d_in[2];  // (N,) float64
  const double* dv = (const double*)d_in[5];  // (K,) float64
  float* out = (float*)d_out;                 // (B, 1) float32
  const int B = in_sizes[0] / KDIM;
  const int blocks = (B + WAVES_PER_BLOCK - 1) / WAVES_PER_BLOCK;
  newsvendor_ip_kernel<<<blocks, 32 * WAVES_PER_BLOCK, 0, stream>>>(
      y, qb, pb, dv, out, B);
}